// PieceWiseActivation_80006650790212
// MI455X (gfx1250) — compile-verified
//
#include <hip/hip_runtime.h>

// Piecewise GELU approximation (hardswish-style), fp32 -> fp32, memory-bound.
//   x <= -3 -> 0
//   x >=  3 -> x
//   else    -> x * (x + 3) / 6
//
// MI455X roofline: 402.7 MB of traffic @ 23.3 TB/s => ~17.3 us floor.
// Strategy: 128-bit non-temporal global loads/stores (data is single-touch,
// in+out = 403 MB > 192 MB L2), grid-stride loop, 256-thread blocks
// (8 wave32 waves).

#define LO (-3.0f)
#define HI (3.0f)
#define INV6 (1.0f / 6.0f)

// native clang vector type: required by __builtin_nontemporal_load/store
typedef __attribute__((ext_vector_type(4))) float v4f;

__device__ __forceinline__ float pw(float x) {
    float mid = x * (x + HI) * INV6;
    float r = (x >= HI) ? x : mid;
    r = (x <= LO) ? 0.0f : r;
    return r;
}

__global__ __launch_bounds__(256) void
piecewise_act_v4_kernel(const v4f* __restrict__ in4,
                        v4f* __restrict__ out4,
                        const float* __restrict__ in,
                        float* __restrict__ out,
                        long long n4, long long n) {
    const long long stride = (long long)gridDim.x * blockDim.x;
    const long long i = (long long)blockIdx.x * blockDim.x + threadIdx.x;

    // main body: 128-bit non-temporal streaming
    for (long long k = i; k < n4; k += stride) {
        v4f v = __builtin_nontemporal_load(&in4[k]);
        v4f r;
        r.x = pw(v.x);
        r.y = pw(v.y);
        r.z = pw(v.z);
        r.w = pw(v.w);
        __builtin_nontemporal_store(r, &out4[k]);
    }

    // scalar tail (n % 4 elements); empty for the reference shape
    for (long long k = n4 * 4 + i; k < n; k += stride) {
        float x = __builtin_nontemporal_load(&in[k]);
        __builtin_nontemporal_store(pw(x), &out[k]);
    }
}

extern "C" void kernel_launch(void* const* d_in, const int* in_sizes, int n_in,
                              void* d_out, int out_size, void* d_ws, size_t ws_size,
                              hipStream_t stream) {
    (void)n_in; (void)d_ws; (void)ws_size; (void)out_size;

    const float* in = (const float*)d_in[0];
    float* out = (float*)d_out;

    const long long n  = (long long)in_sizes[0];
    const long long n4 = n >> 2;

    const int block = 256;
    long long want = (n4 > 0 ? n4 : n);
    long long blocks = (want + block - 1) / block;
    if (blocks < 1) blocks = 1;
    // one float4 per thread covers the whole tensor in a single trip;
    // the grid-stride loop makes any grid size correct.
    if (blocks > 1048576) blocks = 1048576;

    piecewise_act_v4_kernel<<<(dim3)(unsigned)blocks, block, 0, stream>>>(
        (const v4f*)in, (v4f*)out, in, out, n4, n);
}